// SP4CD_P_39178691674666
// MI455X (gfx1250) — compile-verified
//
#include <hip/hip_runtime.h>
#include <math.h>

// Problem constants (from the reference)
#define DIN    384   // D_INNER
#define DSTATE 16
#define DTR    12    // DT_RANK
#define KD     4     // directions
#define NB     2     // batch
#define HH     48
#define LL     2304  // HH*HH
#define CDBL   44    // DT_RANK + 2*D_STATE

typedef float v2f __attribute__((ext_vector_type(2)));
typedef float v8f __attribute__((ext_vector_type(8)));

#if defined(__HIP_DEVICE_COMPILE__)
typedef __attribute__((address_space(1))) int as1_int;   // global
typedef __attribute__((address_space(3))) int as3_int;   // LDS
#endif

// transpose map within the 48x48 grid (involution)
__device__ __forceinline__ int tmap(int l) { return (l % HH) * HH + l / HH; }
// xs[b,k,d,l] = x[b,d, smap(k,l)]
__device__ __forceinline__ int smap(int k, int l) {
    int m = (k >= 2) ? (LL - 1 - l) : l;
    return (k & 1) ? tmap(m) : m;
}

// ---------------------------------------------------------------------------
// Kernel 0: build l-major u buffers for directions 0 (row-major) and 1
// (transposed). Directions 2/3 are read as reversed views of these.
// ---------------------------------------------------------------------------
__global__ void k_build_xst(const float* __restrict__ x, float* __restrict__ xs_t) {
    int tid = blockIdx.x * blockDim.x + threadIdx.x;
    if (tid >= NB * 2 * LL * DIN) return;
    int d   = tid % DIN;
    int l   = (tid / DIN) % LL;
    int dir = (tid / (DIN * LL)) % 2;
    int b   = tid / (DIN * LL * 2);
    xs_t[tid] = x[((size_t)b * DIN + d) * LL + smap(dir, l)];
}

// ---------------------------------------------------------------------------
// Kernel 1: x_dbl[b,k,c,l] = sum_d W[k,c,d] * xs[b,k,d,l]   (44x384 @ 384x2304)
// One wave per 48(M, 3 tiles) x 16(N) block, f32 WMMA 16x16x4, 96 K-steps.
// A (16x4 f32): lane = M + 16*(K>=2), vgpr j -> K = 2*(lane/16)+j  (contiguous
// K-pair => one v2f load). Guard for the ragged M-tile is a hoisted 0/1 mask,
// keeping the inner loop branch-free (EXEC stays all-1s around the WMMAs).
// ---------------------------------------------------------------------------
__global__ void k_xdbl(const float* __restrict__ x, const float* __restrict__ W,
                       float* __restrict__ xdbl) {
    const int nb = blockIdx.x * 16;         // l tile
    const int bk = blockIdx.y;              // b*K + k
    const int k  = bk % KD, b = bk / KD;
    const int ln = threadIdx.x;
    const int half = ln >> 4, lm = ln & 15;
    const float* Wk = W + (size_t)k * CDBL * DIN;
    const float* xb = x + (size_t)b * DIN * LL;
    const int pos = smap(k, nb + lm);       // source column for this lane's N

    const float* wa0 = Wk + (size_t)lm * DIN;
    const float* wa1 = Wk + (size_t)(16 + lm) * DIN;
    const int   c2   = (32 + lm < CDBL) ? (32 + lm) : 0;   // clamped row
    const float m2   = (32 + lm < CDBL) ? 1.0f : 0.0f;     // hoisted mask
    const float* wa2 = Wk + (size_t)c2 * DIN;
    const float* xb0 = xb + pos;

    v8f acc0 = {}, acc1 = {}, acc2 = {};
    for (int kk = 0; kk < DIN / 4; ++kk) {
        const int d0 = kk * 4 + 2 * half;                  // even => 8B aligned
        v2f a0 = *(const v2f*)(wa0 + d0);
        v2f a1 = *(const v2f*)(wa1 + d0);
        v2f a2 = *(const v2f*)(wa2 + d0) * m2;
        v2f bf;
        bf[0] = xb0[(size_t)d0 * LL];
        bf[1] = xb0[(size_t)(d0 + 1) * LL];
        acc0 = __builtin_amdgcn_wmma_f32_16x16x4_f32(false, a0, false, bf, (short)0, acc0, false, false);
        acc1 = __builtin_amdgcn_wmma_f32_16x16x4_f32(false, a1, false, bf, (short)0, acc1, false, false);
        acc2 = __builtin_amdgcn_wmma_f32_16x16x4_f32(false, a2, false, bf, (short)0, acc2, false, false);
    }
    float* out = xdbl + (size_t)bk * CDBL * LL;            // c-major (c, l)
#pragma unroll
    for (int r = 0; r < 8; ++r) {
        int c = r + 8 * half;
        out[(size_t)c * LL + nb + lm]        = acc0[r];
        out[(size_t)(16 + c) * LL + nb + lm] = acc1[r];
        if (32 + c < CDBL) out[(size_t)(32 + c) * LL + nb + lm] = acc2[r];
    }
}

// ---------------------------------------------------------------------------
// Kernel 2: delta[b,k,d,l] = softplus( dtw(384x12) @ dts(12x2304) + bias )
// One wave per 16x16 tile, 3 WMMA K-steps. Output stored l-major for the scan.
// ---------------------------------------------------------------------------
__global__ void k_delta(const float* __restrict__ xdbl, const float* __restrict__ dtw,
                        const float* __restrict__ bias, float* __restrict__ delta_t) {
    const int nb = blockIdx.x * 16;   // l tile
    const int mb = blockIdx.y * 16;   // d tile
    const int bk = blockIdx.z;
    const int k  = bk % KD;
    const int ln = threadIdx.x, half = ln >> 4, lm = ln & 15;
    const float* Ak = dtw + (size_t)k * DIN * DTR;
    const float* Bp = xdbl + (size_t)bk * CDBL * LL;   // dts rows are c = 0..11
    v8f acc = {};
#pragma unroll
    for (int kk = 0; kk < 3; ++kk) {
        v2f a = *(const v2f*)&Ak[(size_t)(mb + lm) * DTR + kk * 4 + 2 * half];
        v2f bfr;
#pragma unroll
        for (int j = 0; j < 2; ++j) {
            int r = kk * 4 + 2 * half + j;
            bfr[j] = Bp[(size_t)r * LL + nb + lm];
        }
        acc = __builtin_amdgcn_wmma_f32_16x16x4_f32(false, a, false, bfr, (short)0, acc, false, false);
    }
#pragma unroll
    for (int r = 0; r < 8; ++r) {
        int d = mb + r + 8 * half;
        float v = acc[r] + bias[k * DIN + d];
        v = (v > 20.0f) ? v : log1pf(__expf(v));               // softplus
        delta_t[((size_t)bk * LL + nb + lm) * DIN + d] = v;     // l-major
    }
}

// ---------------------------------------------------------------------------
// Kernel 3: selective scan. One block per (b,k); thread d owns h[16] in regs.
// B/C (shared across all 384 channels) are staged into LDS per 96-step chunk
// using gfx1250 async VMEM->LDS copies (ASYNCcnt) when available, then read as
// lane-broadcasts. delta/u/y accesses are coalesced (l-major); next chunk's
// delta/u streams are prefetched while the current chunk computes.
// ---------------------------------------------------------------------------
#define CH 96
__global__ void k_scan(const float* __restrict__ xdbl, const float* __restrict__ delta_t,
                       const float* __restrict__ xs_t, const float* __restrict__ A_logs,
                       const float* __restrict__ Ds, float* __restrict__ ydir) {
    __shared__ float sh[32 * CH];   // sh[i*CH + step]; i<16: B[n], i>=16: C[n]
    const int bk = blockIdx.x;
    const int k = bk % KD, b = bk / KD;
    const int d = threadIdx.x;
    float A[DSTATE], h[DSTATE];
#pragma unroll
    for (int n = 0; n < DSTATE; ++n) {
        A[n] = -__expf(A_logs[(size_t)(k * DIN + d) * DSTATE + n]);
        h[n] = 0.0f;
    }
    const float Dv = Ds[k * DIN + d];
    const float* bc = xdbl + ((size_t)bk * CDBL + DTR) * LL;            // Bs,Cs rows
    const float* dl = delta_t + (size_t)bk * LL * DIN;
    const float* us = xs_t + (size_t)(b * 2 + (k & 1)) * LL * DIN;      // dirs 2,3 reuse reversed
    float* yo = ydir + (size_t)bk * LL * DIN;

    for (int l0 = 0; l0 < LL; l0 += CH) {
        __syncthreads();   // previous chunk fully consumed before overwrite
        for (int idx = threadIdx.x; idx < 32 * CH; idx += DIN) {
            int i = idx / CH, st = idx % CH;                            // coalesced along l
#if __has_builtin(__builtin_amdgcn_global_load_async_to_lds_b32)
            __builtin_amdgcn_global_load_async_to_lds_b32(
                (as1_int*)&bc[(size_t)i * LL + l0 + st],
                (as3_int*)&sh[idx], 0, 0);
#else
            sh[idx] = bc[(size_t)i * LL + l0 + st];
#endif
        }
#if __has_builtin(__builtin_amdgcn_global_load_async_to_lds_b32)
#if __has_builtin(__builtin_amdgcn_s_wait_asynccnt)
        __builtin_amdgcn_s_wait_asynccnt(0);
#else
        asm volatile("s_wait_asynccnt 0x0" ::: "memory");
#endif
#endif
        __syncthreads();

        // pull next chunk's per-channel streams toward the WGP while we compute
        if (l0 + CH < LL) {
            int lnx = l0 + CH;
            int lux = (k < 2) ? lnx : (LL - 1 - lnx - (CH - 1));
            __builtin_prefetch(&dl[(size_t)lnx * DIN + d], 0, 1);
            __builtin_prefetch(&us[(size_t)lux * DIN + d], 0, 1);
        }

        for (int ls = 0; ls < CH; ++ls) {
            int l  = l0 + ls;
            int lu = (k < 2) ? l : (LL - 1 - l);
            float u  = us[(size_t)lu * DIN + d];
            float dt = dl[(size_t)l * DIN + d];
            float du = dt * u;
            float y = 0.0f;
#pragma unroll
            for (int n = 0; n < DSTATE; ++n) {
                float dA = __expf(dt * A[n]);
                h[n] = dA * h[n] + du * sh[n * CH + ls];
                y   += h[n] * sh[(DSTATE + n) * CH + ls];
            }
            yo[(size_t)l * DIN + d] = y + Dv * u;
        }
    }
}

// ---------------------------------------------------------------------------
// Kernel 4: recombine 4 directions into out[b,h,w,d] (fully overwrites d_out).
// ---------------------------------------------------------------------------
__global__ void k_combine(const float* __restrict__ ydir, float* __restrict__ out) {
    int tid = blockIdx.x * blockDim.x + threadIdx.x;
    if (tid >= NB * LL * DIN) return;
    int d = tid % DIN;
    int p = (tid / DIN) % LL;
    int b = tid / (DIN * LL);
    int pt = tmap(p);
    const float* y = ydir + (size_t)b * KD * LL * DIN;
    float v = y[((size_t)0 * LL + p) * DIN + d]
            + y[((size_t)1 * LL + pt) * DIN + d]
            + y[((size_t)2 * LL + (LL - 1 - p)) * DIN + d]
            + y[((size_t)3 * LL + (LL - 1 - pt)) * DIN + d];
    out[tid] = v;
}

// ---------------------------------------------------------------------------
extern "C" void kernel_launch(void* const* d_in, const int* in_sizes, int n_in,
                              void* d_out, int out_size, void* d_ws, size_t ws_size,
                              hipStream_t stream) {
    const float* x      = (const float*)d_in[0];   // (B, 384, 48, 48)
    const float* xpw    = (const float*)d_in[1];   // (K, 44, 384)
    const float* dtw    = (const float*)d_in[2];   // (K, 384, 12)
    const float* dtb    = (const float*)d_in[3];   // (K, 384)
    const float* A_logs = (const float*)d_in[4];   // (K*384, 16)
    const float* Ds     = (const float*)d_in[5];   // (K*384,)
    float* out = (float*)d_out;                    // (B, 48, 48, 384)

    const size_t n_xdbl  = (size_t)NB * KD * CDBL * LL;  // 3.2 MB
    const size_t n_delta = (size_t)NB * KD * LL * DIN;   // 28.3 MB
    const size_t n_xst   = (size_t)NB * 2 * LL * DIN;    // 14.2 MB
    float* ws      = (float*)d_ws;
    float* xdbl    = ws;
    float* delta_t = xdbl + n_xdbl;
    float* xs_t    = delta_t + n_delta;
    float* ydir    = xs_t + n_xst;                       // 28.3 MB; total ~74 MB

    (void)in_sizes; (void)n_in; (void)out_size; (void)ws_size;

    k_build_xst<<<(int)((n_xst + 255) / 256), 256, 0, stream>>>(x, xs_t);
    k_xdbl<<<dim3(LL / 16, NB * KD), 32, 0, stream>>>(x, xpw, xdbl);
    k_delta<<<dim3(LL / 16, DIN / 16, NB * KD), 32, 0, stream>>>(xdbl, dtw, dtb, delta_t);
    k_scan<<<NB * KD, DIN, 0, stream>>>(xdbl, delta_t, xs_t, A_logs, Ds, ydir);
    k_combine<<<(int)(((size_t)NB * LL * DIN + 255) / 256), 256, 0, stream>>>(ydir, out);
}